// MtpDrafterANE_47399259078790
// MI455X (gfx1250) — compile-verified
//
#include <hip/hip_runtime.h>
#include <math.h>

typedef float v2f __attribute__((ext_vector_type(2)));
typedef float v8f __attribute__((ext_vector_type(8)));
typedef int   v4i __attribute__((ext_vector_type(4)));

#define WAVE 32
#define CST 132   // padded LDS row stride per 128-col chunk: 4-float aligned, A-frag reads conflict-free

// ---- gfx1250 async global->LDS path (ASYNCcnt), with safe fallback --------
#if defined(__gfx1250__) && __has_builtin(__builtin_amdgcn_global_load_async_to_lds_b128) && \
    __has_builtin(__builtin_amdgcn_s_wait_asynccnt)
#define HAVE_ASYNC_LDS 1
#else
#define HAVE_ASYNC_LDS 0
#endif

__device__ __forceinline__ void async_cp16(const float* g, float* l) {
#if HAVE_ASYNC_LDS
  // builtin signature (from probe): (v4i addrspace(1)*, v4i addrspace(3)*, imm offset, imm cpol)
  unsigned long long ga = (unsigned long long)g;
  unsigned int       la = (unsigned int)(unsigned long long)l;  // generic LDS ptr: addr[31:0] = LDS byte addr
  __builtin_amdgcn_global_load_async_to_lds_b128(
      (__attribute__((address_space(1))) v4i*)ga,
      (__attribute__((address_space(3))) v4i*)la, 0, 0);
#else
  *(float4*)l = *(const float4*)g;
#endif
}

#if HAVE_ASYNC_LDS
#define WAIT_ASYNC(n) __builtin_amdgcn_s_wait_asynccnt(n)
#else
#define WAIT_ASYNC(n)
#endif

__device__ __forceinline__ float wave_sum(float v) {
  for (int off = 16; off; off >>= 1) v += __shfl_xor(v, off, 32);
  return v;
}

// ---------------- generic GEMV: y[r] = dot(W[rowidx ? rowidx[r] : r], x) ----------------
__global__ void gemv_kernel(const float* __restrict__ W, const float* __restrict__ x,
                            float* __restrict__ y, int R, int C,
                            const int* __restrict__ rowidx) {
  int wid = threadIdx.x >> 5, lane = threadIdx.x & 31;
  int r = blockIdx.x * (blockDim.x >> 5) + wid;
  if (r >= R) return;
  size_t row = rowidx ? (size_t)rowidx[r] : (size_t)r;
  const float* Wr = W + row * (size_t)C;
  float acc = 0.f;
  for (int i = lane * 4; i < C; i += 128) {
    __builtin_prefetch(Wr + i + 512, 0, 0);
    float4 wv = *(const float4*)(Wr + i);
    float4 xv = *(const float4*)(x + i);
    acc += wv.x * xv.x + wv.y * xv.y + wv.z * xv.z + wv.w * xv.w;
  }
  acc = wave_sum(acc);
  if (lane == 0) y[r] = acc;
}

// ---------------- fused dual GEMV for input projection ----------------
__global__ void gemv2_kernel(const float* __restrict__ W1, const float* __restrict__ x1,
                             const float* __restrict__ W2, const float* __restrict__ x2,
                             float* __restrict__ y, int R, int C) {
  int wid = threadIdx.x >> 5, lane = threadIdx.x & 31;
  int r = blockIdx.x * (blockDim.x >> 5) + wid;
  if (r >= R) return;
  const float* W1r = W1 + (size_t)r * C;
  const float* W2r = W2 + (size_t)r * C;
  float acc = 0.f;
  for (int i = lane * 4; i < C; i += 128) {
    float4 a = *(const float4*)(W1r + i);
    float4 b = *(const float4*)(x1 + i);
    float4 c = *(const float4*)(W2r + i);
    float4 d = *(const float4*)(x2 + i);
    acc += a.x * b.x + a.y * b.y + a.z * b.z + a.w * b.w;
    acc += c.x * d.x + c.y * d.y + c.z * d.z + c.w * d.w;
  }
  acc = wave_sum(acc);
  if (lane == 0) y[r] = acc;
}

// ---------------- RMSNorm (single block): out = (res?res:0) + in*rsqrt(mean(in^2)+eps)*w ----------------
__global__ void rmsnorm_kernel(float* __restrict__ out, const float* __restrict__ in,
                               const float* __restrict__ w, int n,
                               const float* __restrict__ res) {
  __shared__ float red[32];
  float ss = 0.f;
  for (int i = threadIdx.x; i < n; i += blockDim.x) { float v = in[i]; ss += v * v; }
  ss = wave_sum(ss);
  int wid = threadIdx.x >> 5, lane = threadIdx.x & 31;
  if (lane == 0) red[wid] = ss;
  __syncthreads();
  if (threadIdx.x == 0) {
    float t = 0.f;
    for (int i = 0; i < (int)(blockDim.x >> 5); i++) t += red[i];
    red[0] = t;
  }
  __syncthreads();
  float rms = rsqrtf(red[0] / (float)n + 1e-6f);
  for (int i = threadIdx.x; i < n; i += blockDim.x) {
    float v = in[i] * rms * w[i];
    out[i] = res ? (res[i] + v) : v;
  }
}

// ---------------- per-head q RMSNorm + inline RoPE (8 heads x 256, one wave/head) ----------------
__global__ void qnorm_rope_kernel(float* __restrict__ q, const float* __restrict__ qw,
                                  const float* __restrict__ cs, const float* __restrict__ sn) {
  int wid = threadIdx.x >> 5, lane = threadIdx.x & 31;
  float* qh = q + wid * 256;
  float v[8]; float ss = 0.f;
  for (int j = 0; j < 8; j++) { int d = lane + j * 32; v[j] = qh[d]; ss += v[j] * v[j]; }
  ss = wave_sum(ss);
  float rms = rsqrtf(ss / 256.f + 1e-6f);
  float outv[8];
  for (int j = 0; j < 8; j++) {
    int d = lane + j * 32;
    float qv = v[j] * rms * qw[d];
    int dr = (d < 128) ? d + 128 : d - 128;
    float rq = qh[dr] * rms * qw[dr];
    float rot = (d < 128) ? -rq : rq;
    outv[j] = qv * cs[d & 127] + rot * sn[d & 127];
  }
  __syncthreads();
  for (int j = 0; j < 8; j++) { int d = lane + j * 32; qh[d] = outv[j]; }
}

// ---------------- attention scores via fp32 WMMA 16x16x4, async double-buffered K staging ----
// A = K tile (16 s-rows x 4 d), B = q broadcast over 16 columns, D row m = score(s_base+m).
// One wave per (s-tile, head). K tile (16x256 f32 = 16KB) split into two 128-col chunks;
// both issued as GLOBAL_LOAD_ASYNC_TO_LDS_B128, chunk-1 HBM latency overlapped with
// chunk-0 WMMA work via s_wait_asynccnt 16 / 0.
__device__ __forceinline__ v8f score_chunk(const float* kb, const float* qb, int lane, v8f acc) {
  int m = lane & 15;
  int koff = (lane >> 4) * 2;   // lanes 0-15: K=0,1 ; lanes 16-31: K=2,3
  for (int d = 0; d < 128; d += 4) {
    v2f a, b;
    a.x = kb[m * CST + d + koff];
    a.y = kb[m * CST + d + koff + 1];
    b.x = qb[d + koff];
    b.y = qb[d + koff + 1];
    acc = __builtin_amdgcn_wmma_f32_16x16x4_f32(false, a, false, b, (short)0, acc, false, false);
  }
  return acc;
}

__global__ void attn_scores_kernel(const float* __restrict__ K, const float* __restrict__ q,
                                   const float* __restrict__ mask, float* __restrict__ scores) {
  __shared__ float kbuf[2][16 * CST];
  __shared__ float qrow[256];
  int lane = threadIdx.x & 31;
  int stile = blockIdx.x, head = blockIdx.y;
  const float* Kb = K + ((size_t)head * 4096 + (size_t)stile * 16) * 256;

  for (int j = 0; j < 8; j++) qrow[lane + j * 32] = q[head * 256 + lane + j * 32];

  // issue both chunks (16 x b128 per chunk per wave, coalesced 128B lines)
  for (int r = 0; r < 16; r++)
    async_cp16(Kb + r * 256 + lane * 4, &kbuf[0][r * CST + lane * 4]);
  for (int r = 0; r < 16; r++)
    async_cp16(Kb + r * 256 + 128 + lane * 4, &kbuf[1][r * CST + lane * 4]);

  v8f acc = {};
  WAIT_ASYNC(16);          // first 16 async transfers (chunk 0) complete, in-order
  __syncthreads();
  acc = score_chunk(kbuf[0], qrow, lane, acc);        // overlaps chunk-1 HBM fetch
  WAIT_ASYNC(0);
  __syncthreads();
  acc = score_chunk(kbuf[1], qrow + 128, lane, acc);

  // D: VGPR j holds row m=j (lanes 0-15) / m=j+8 (lanes 16-31); all 16 columns identical.
  if ((lane & 15) == 0) {
    int mbase = (lane >> 4) * 8;
    for (int j = 0; j < 8; j++) {
      int s = stile * 16 + mbase + j;
      scores[(size_t)head * 4096 + s] = acc[j] + mask[s];
    }
  }
}

// ---------------- row softmax (one block per head) ----------------
__global__ void softmax_kernel(float* __restrict__ scores, int n) {
  __shared__ float red[32];
  float* row = scores + (size_t)blockIdx.x * n;
  int wid = threadIdx.x >> 5, lane = threadIdx.x & 31;
  int nw = blockDim.x >> 5;
  float mx = -3.4e38f;
  for (int i = threadIdx.x; i < n; i += blockDim.x) mx = fmaxf(mx, row[i]);
  for (int off = 16; off; off >>= 1) mx = fmaxf(mx, __shfl_xor(mx, off, 32));
  if (lane == 0) red[wid] = mx;
  __syncthreads();
  if (threadIdx.x == 0) { float t = red[0]; for (int i = 1; i < nw; i++) t = fmaxf(t, red[i]); red[0] = t; }
  __syncthreads();
  mx = red[0];
  __syncthreads();
  float s = 0.f;
  for (int i = threadIdx.x; i < n; i += blockDim.x) { float e = expf(row[i] - mx); row[i] = e; s += e; }
  s = wave_sum(s);
  if (lane == 0) red[wid] = s;
  __syncthreads();
  if (threadIdx.x == 0) { float t = 0.f; for (int i = 0; i < nw; i++) t += red[i]; red[0] = t; }
  __syncthreads();
  float inv = 1.f / red[0];
  for (int i = threadIdx.x; i < n; i += blockDim.x) row[i] *= inv;
}

// ---------------- o[n,d] = sum_s attn[n,s] * V[n,d,s]  (V stored [NH,HD,S], contiguous s) ----------------
__global__ void attnv_kernel(const float* __restrict__ V, const float* __restrict__ attn,
                             float* __restrict__ o) {
  int wid = threadIdx.x >> 5, lane = threadIdx.x & 31;
  int r = blockIdx.x * 8 + wid;            // 0..2047 = head*256 + d
  int n = r >> 8, d = r & 255;
  const float* Vr = V + ((size_t)n * 256 + d) * 4096;
  const float* Ar = attn + (size_t)n * 4096;
  float acc = 0.f;
  for (int s = lane * 4; s < 4096; s += 128) {
    float4 vv = *(const float4*)(Vr + s);
    float4 av = *(const float4*)(Ar + s);
    acc += vv.x * av.x + vv.y * av.y + vv.z * av.z + vv.w * av.w;
  }
  acc = wave_sum(acc);
  if (lane == 0) o[r] = acc;
}

// ---------------- gelu(tanh) * u ----------------
__global__ void gelu_mul_kernel(const float* __restrict__ g, const float* __restrict__ u,
                                float* __restrict__ a, int n) {
  int i = blockIdx.x * blockDim.x + threadIdx.x;
  if (i < n) {
    float x = g[i];
    float t = tanhf(0.7978845608f * (x + 0.044715f * x * x * x));
    a[i] = 0.5f * x * (1.f + t) * u[i];
  }
}

// ---------------- iterative block top-k over n<=2048 ----------------
__global__ void topk_kernel(const float* __restrict__ vals, int n, int k,
                            int* __restrict__ oidx, float* __restrict__ oval) {
  __shared__ float buf[2048];
  __shared__ float rv[256];
  __shared__ int ri[256];
  for (int i = threadIdx.x; i < n; i += blockDim.x) buf[i] = vals[i];
  __syncthreads();
  for (int t = 0; t < k; t++) {
    float bv = -3.4e38f; int bi = 0;
    for (int i = threadIdx.x; i < n; i += blockDim.x)
      if (buf[i] > bv) { bv = buf[i]; bi = i; }
    rv[threadIdx.x] = bv; ri[threadIdx.x] = bi;
    __syncthreads();
    for (int off = blockDim.x >> 1; off; off >>= 1) {
      if ((int)threadIdx.x < off) {
        if (rv[threadIdx.x + off] > rv[threadIdx.x]) {
          rv[threadIdx.x] = rv[threadIdx.x + off];
          ri[threadIdx.x] = ri[threadIdx.x + off];
        }
      }
      __syncthreads();
    }
    if (threadIdx.x == 0) { oidx[t] = ri[0]; oval[t] = rv[0]; buf[ri[0]] = -3.4e38f; }
    __syncthreads();
  }
}

// ---------------- sel[j] = token_ordering[cluster[j/64]*64 + j%64] ----------------
__global__ void build_sel_kernel(const int* __restrict__ tord, const int* __restrict__ cidx,
                                 int* __restrict__ sel) {
  int j = blockIdx.x * blockDim.x + threadIdx.x;
  if (j < 2048) sel[j] = tord[cidx[j >> 6] * 64 + (j & 63)];
}

// ---------------- final: ids (as float) + vals into out[0..15] ----------------
__global__ void final_out_kernel(const int* __restrict__ sel, const int* __restrict__ tpos,
                                 const float* __restrict__ tval, float* __restrict__ out) {
  int i = threadIdx.x;
  if (i < 8) { out[i] = (float)sel[tpos[i]]; out[8 + i] = tval[i]; }
}

// ---------------- workspace layout (floats) ----------------
#define WS_X     0
#define WS_H     2048
#define WS_Q     4096
#define WS_TMP   6144
#define WS_O     8192
#define WS_SC    10240          // 8*4096 = 32768
#define WS_G     43008
#define WS_U     51200
#define WS_ACT   59392
#define WS_HV    67584
#define WS_CLOG  69632
#define WS_CIDX  71680          // 32 ints
#define WS_CVAL  71712          // 32 floats
#define WS_SEL   71744          // 2048 ints
#define WS_SELOG 73792          // 2048 floats
#define WS_TPOS  75840          // 8 ints
#define WS_TVAL  75848          // 8 floats

extern "C" void kernel_launch(void* const* d_in, const int* in_sizes, int n_in,
                              void* d_out, int out_size, void* d_ws, size_t ws_size,
                              hipStream_t stream) {
  (void)in_sizes; (void)n_in; (void)out_size; (void)ws_size;

  const float* emb    = (const float*)d_in[0];
  const float* pact   = (const float*)d_in[1];
  const float* kvk[2]  = { (const float*)d_in[2], (const float*)d_in[4] };
  const float* kvv[2]  = { (const float*)d_in[3], (const float*)d_in[5] };
  const float* cosv[2] = { (const float*)d_in[6], (const float*)d_in[8] };
  const float* sinv[2] = { (const float*)d_in[7], (const float*)d_in[9] };
  const float* mskv[2] = { (const float*)d_in[10], (const float*)d_in[11] };
  const float* w_in_e = (const float*)d_in[12];
  const float* w_in_p = (const float*)d_in[13];
  const float* fnw    = (const float*)d_in[14];
  const float* w_post = (const float*)d_in[15];
  const float* lmhead = (const float*)d_in[16];
  const float* cent   = (const float*)d_in[17];
  const int*   tord   = (const int*)d_in[18];

  float* ws  = (float*)d_ws;
  float* out = (float*)d_out;
  float* X = ws + WS_X;   float* H = ws + WS_H;   float* Q = ws + WS_Q;
  float* T = ws + WS_TMP; float* O = ws + WS_O;   float* SC = ws + WS_SC;
  float* G = ws + WS_G;   float* U = ws + WS_U;   float* ACT = ws + WS_ACT;
  float* HV = ws + WS_HV; float* CLOG = ws + WS_CLOG;
  int*   CIDX = (int*)(ws + WS_CIDX);  float* CVAL = ws + WS_CVAL;
  int*   SEL  = (int*)(ws + WS_SEL);   float* SELOG = ws + WS_SELOG;
  int*   TPOS = (int*)(ws + WS_TPOS);  float* TVAL = ws + WS_TVAL;

  // x = w_in_e @ e + w_in_p @ p
  gemv2_kernel<<<256, 256, 0, stream>>>(w_in_e, emb, w_in_p, pact, X, 2048, 2048);

  for (int l = 0; l < 2; l++) {
    int pb = 19 + 10 * l;
    const float* ln_in = (const float*)d_in[pb + 0];
    const float* ln_pa = (const float*)d_in[pb + 1];
    const float* ln_pf = (const float*)d_in[pb + 2];
    const float* ln_pof= (const float*)d_in[pb + 3];
    const float* qn    = (const float*)d_in[pb + 4];
    const float* wq    = (const float*)d_in[pb + 5];
    const float* wo    = (const float*)d_in[pb + 6];
    const float* wg    = (const float*)d_in[pb + 7];
    const float* wu    = (const float*)d_in[pb + 8];
    const float* wd    = (const float*)d_in[pb + 9];

    rmsnorm_kernel<<<1, 256, 0, stream>>>(H, X, ln_in, 2048, nullptr);
    gemv_kernel<<<256, 256, 0, stream>>>(wq, H, Q, 2048, 2048, nullptr);
    qnorm_rope_kernel<<<1, 256, 0, stream>>>(Q, qn, cosv[l], sinv[l]);
    attn_scores_kernel<<<dim3(256, 8), 32, 0, stream>>>(kvk[l], Q, mskv[l], SC);
    softmax_kernel<<<8, 256, 0, stream>>>(SC, 4096);
    attnv_kernel<<<256, 256, 0, stream>>>(kvv[l], SC, O);
    gemv_kernel<<<256, 256, 0, stream>>>(wo, O, T, 2048, 2048, nullptr);
    rmsnorm_kernel<<<1, 256, 0, stream>>>(X, T, ln_pa, 2048, X);       // x = x + rms(wo@o)
    rmsnorm_kernel<<<1, 256, 0, stream>>>(H, X, ln_pf, 2048, nullptr);
    gemv_kernel<<<1024, 256, 0, stream>>>(wg, H, G, 8192, 2048, nullptr);
    gemv_kernel<<<1024, 256, 0, stream>>>(wu, H, U, 8192, 2048, nullptr);
    gelu_mul_kernel<<<32, 256, 0, stream>>>(G, U, ACT, 8192);
    gemv_kernel<<<256, 256, 0, stream>>>(wd, ACT, T, 2048, 8192, nullptr);
    rmsnorm_kernel<<<1, 256, 0, stream>>>(X, T, ln_pof, 2048, X);      // x = x + rms(mlp)
  }

  rmsnorm_kernel<<<1, 256, 0, stream>>>(HV, X, fnw, 2048, nullptr);
  gemv_kernel<<<256, 256, 0, stream>>>(cent, HV, CLOG, 2048, 2048, nullptr);
  topk_kernel<<<1, 256, 0, stream>>>(CLOG, 2048, 32, CIDX, CVAL);
  build_sel_kernel<<<8, 256, 0, stream>>>(tord, CIDX, SEL);
  gemv_kernel<<<256, 256, 0, stream>>>(lmhead, HV, SELOG, 2048, 2048, SEL);
  topk_kernel<<<1, 256, 0, stream>>>(SELOG, 2048, 8, TPOS, TVAL);
  gemv_kernel<<<256, 256, 0, stream>>>(w_post, HV, out + 16, 2048, 2048, nullptr);
  final_out_kernel<<<1, 32, 0, stream>>>(SEL, TPOS, TVAL, out);
}